// _delayLayer_2576980377764
// MI455X (gfx1250) — compile-verified
//
#include <hip/hip_runtime.h>
#include <hip/hip_bf16.h>

// Problem geometry (fixed by the reference): (N,C,H,W,T) = (8,16,32,32,500)
#define T_LEN            500
#define T4               125                    // float4s per row (500/4)
#define THREADS_PER_ROW  128                    // 4 waves cover 125 float4s
#define ROWS_PER_BLOCK   2
#define BLOCK_THREADS    (ROWS_PER_BLOCK * THREADS_PER_ROW)
#define SMEM_STRIDE      (T_LEN + 4)            // pad; 504*4B = 2016B keeps 16B alignment

typedef __attribute__((ext_vector_type(4))) float f4;
typedef __attribute__((ext_vector_type(4))) int   v4i;

// ---- gfx1250 async global->LDS path (guarded; fallback = b128 load + ds_store) ----
#if defined(__has_builtin)
#  if __has_builtin(__builtin_amdgcn_global_load_async_to_lds_b128)
#    define USE_ASYNC_LDS 1
#  endif
#  if __has_builtin(__builtin_amdgcn_s_wait_asynccnt)
#    define HAVE_WAIT_ASYNC_BUILTIN 1
#  endif
#endif

#if defined(__AMDGCN__) && !defined(USE_ASYNC_LDS)
#warning "gfx1250 async-to-LDS builtin not available on this toolchain; using ds_store fallback"
#endif

#ifdef __AMDGCN__
typedef __attribute__((address_space(1))) v4i gv4i_t;   // global 4xi32
typedef __attribute__((address_space(3))) v4i lv4i_t;   // LDS    4xi32
#endif

__launch_bounds__(BLOCK_THREADS)
__global__ void delay_shift_kernel(const float* __restrict__ in,
                                   const float* __restrict__ delay,
                                   const int*   __restrict__ ts_ptr,
                                   float*       __restrict__ out,
                                   int n_rows, int chw, int chw_mask) {
  __shared__ float smem[ROWS_PER_BLOCK][SMEM_STRIDE];

  const int tid  = threadIdx.x;
  const int rsub = tid / THREADS_PER_ROW;          // which row within the block
  const int lane = tid % THREADS_PER_ROW;          // 0..127 within the row
  // `row` is uniform across each wave (128 threads/row = 4 full wave32s);
  // force it scalar so per-row parameters use SALU/SMEM instead of VALU.
  const int row  = __builtin_amdgcn_readfirstlane(
                       blockIdx.x * ROWS_PER_BLOCK + rsub);
  const bool rowValid = (row < n_rows);
  const bool active   = rowValid && (lane < T4);

  // ---- Stage one full row (500 floats) into LDS with 128-bit transfers ----
  if (active) {
    const float* gp = in + (size_t)row * T_LEN + (size_t)lane * 4;
    float*       lp = &smem[rsub][lane * 4];
#ifdef USE_ASYNC_LDS
    __builtin_amdgcn_global_load_async_to_lds_b128((gv4i_t*)gp, (lv4i_t*)lp,
                                                   /*offset=*/0, /*cpol=*/0);
#else
    *(f4*)lp = *(const f4*)gp;
#endif
  }

#ifdef USE_ASYNC_LDS
  // Each wave drains its own async-copy counter; the workgroup barrier does
  // NOT cover ASYNCcnt-tracked LDS writes.
#  ifdef HAVE_WAIT_ASYNC_BUILTIN
  __builtin_amdgcn_s_wait_asynccnt(0);
#  else
  asm volatile("s_wait_asynccnt 0" ::: "memory");
#  endif
#endif
  __syncthreads();

  if (!active) return;

  // ---- Per-row shift parameters (wave-uniform -> scalar pipe) ----
  const int chwIdx = (chw_mask >= 0) ? (row & chw_mask)   // power-of-two fast path
                                     : (row % chw);
  const float Ts   = (float)ts_ptr[0];
  const float s    = delay[chwIdx] / Ts;
  const float sif  = floorf(s);
  const float frac = s - sif;
  const int   si   = (int)sif;

  // For outputs t0..t0+3 we need x[t0-si-1 .. t0+3-si]: 5 contiguous values.
  const float* srow = smem[rsub];
  const int t0   = lane * 4;
  const int base = t0 - si;

  float v[5];
#pragma unroll
  for (int k = 0; k < 5; ++k) {
    const int idx = base - 1 + k;
    v[k] = ((unsigned)idx < (unsigned)T_LEN) ? srow[idx] : 0.0f;
  }

  const float w0 = 1.0f - frac;
  f4 r;
  r.x = w0 * v[1] + frac * v[0];
  r.y = w0 * v[2] + frac * v[1];
  r.z = w0 * v[3] + frac * v[2];
  r.w = w0 * v[4] + frac * v[3];

  // Coalesced 128-bit store of the finished row segment.
  ((f4*)out)[(size_t)row * T4 + lane] = r;
}

extern "C" void kernel_launch(void* const* d_in, const int* in_sizes, int n_in,
                              void* d_out, int out_size, void* d_ws, size_t ws_size,
                              hipStream_t stream) {
  const float* in    = (const float*)d_in[0];   // (N,C,H,W,T) fp32
  const float* delay = (const float*)d_in[1];   // (C,H,W)     fp32
  const int*   ts    = (const int*)d_in[2];     // scalar Ts (int)
  float*       out   = (float*)d_out;

  const int total  = in_sizes[0];               // N*C*H*W*T
  const int chw    = in_sizes[1];               // C*H*W
  const int n_rows = total / T_LEN;             // N*C*H*W
  const int blocks = (n_rows + ROWS_PER_BLOCK - 1) / ROWS_PER_BLOCK;
  // chw = 16384 here: power of two -> replace the per-thread integer modulo
  // with a mask; keep a % fallback for generality.
  const int chw_mask = ((chw & (chw - 1)) == 0) ? (chw - 1) : -1;

  delay_shift_kernel<<<blocks, BLOCK_THREADS, 0, stream>>>(in, delay, ts, out,
                                                           n_rows, chw, chw_mask);
}